// AtomCrossAttEncoder_90812788507390
// MI455X (gfx1250) — compile-verified
//
#include <hip/hip_runtime.h>
#include <hip/hip_bf16.h>

// ---------------------------------------------------------------------------
// AtomCrossAttEncoder for MI455X (gfx1250, wave32).
// f32 end-to-end; all 16-channel projections/MLP on V_WMMA_F32_16X16X4_F32;
// trunk_pair LN+proj streams through async global->LDS copies.
// ---------------------------------------------------------------------------

typedef __attribute__((ext_vector_type(2))) float v2f;
typedef __attribute__((ext_vector_type(8))) float v8f;

__device__ __forceinline__ v8f wmma_f32_4(v2f a, v2f b, v8f c) {
  // D = A(16x4 f32) x B(4x16 f32) + C(16x16 f32)
  return __builtin_amdgcn_wmma_f32_16x16x4_f32(false, a, false, b, (short)0, c,
                                               false, false);
}

#define T_TOK 768
#define A_ATM 24
#define S_SUB 576
#define Q_NUM 32
#define K_NUM 128
#define CA 128
#define CP 16
#define TPC_PAD 132   // LDS row stride in floats (128 + 4 pad -> conflict-free)

// ---------------------------------------------------------------------------
// Kernel 1: per-atom conditioning on (T,A) layout -> tasc (T*A, CA)
// ---------------------------------------------------------------------------
__global__ void __launch_bounds__(128)
k_atom(const float* __restrict__ ref_ops, const float* __restrict__ ref_mask,
       const int* __restrict__ ref_elem, const float* __restrict__ ref_chg,
       const int* __restrict__ ref_name,
       const float* __restrict__ w_pos, const float* __restrict__ w_maskw,
       const float* __restrict__ w_elem, const float* __restrict__ w_chg,
       const float* __restrict__ w_name, float* __restrict__ tasc)
{
  const int row = blockIdx.x;     // t*A + a, 0..18431
  const int c   = threadIdx.x;    // 0..127
  const float p0 = ref_ops[row * 3 + 0];
  const float p1 = ref_ops[row * 3 + 1];
  const float p2 = ref_ops[row * 3 + 2];
  const float mm = ref_mask[row];
  float a = p0 * w_pos[0 * CA + c] + p1 * w_pos[1 * CA + c] + p2 * w_pos[2 * CA + c];
  a += mm * w_maskw[c];
  a += w_elem[(long)ref_elem[row] * CA + c];
  a += asinhf(ref_chg[row]) * w_chg[c];
#pragma unroll
  for (int i = 0; i < 4; ++i)
    a += w_name[(long)(i * 64 + ref_name[row * 4 + i]) * CA + c];
  tasc[(long)row * CA + c] = a * mm;
}

// ---------------------------------------------------------------------------
// Kernel 2: tsc = LN(trunk_single_cond) @ w_ts  -> (768, 128)
// ---------------------------------------------------------------------------
__global__ void __launch_bounds__(128)
k_tsc(const float* __restrict__ x, const float* __restrict__ scale,
      const float* __restrict__ w, float* __restrict__ out)
{
  __shared__ float sx[384];
  __shared__ float rs[128], rs2[128];
  const int t = blockIdx.x, tid = threadIdx.x;
  const float* xr = x + (long)t * 384;
  for (int i = tid; i < 384; i += 128) sx[i] = xr[i];
  __syncthreads();
  float s = 0.f, s2 = 0.f;
  for (int i = tid; i < 384; i += 128) { float v = sx[i]; s += v; s2 += v * v; }
  rs[tid] = s; rs2[tid] = s2;
  __syncthreads();
  for (int o = 64; o; o >>= 1) {
    if (tid < o) { rs[tid] += rs[tid + o]; rs2[tid] += rs2[tid + o]; }
    __syncthreads();
  }
  const float mu  = rs[0] * (1.f / 384.f);
  const float var = rs2[0] * (1.f / 384.f) - mu * mu;
  const float r   = rsqrtf(var + 1e-5f);
  for (int i = tid; i < 384; i += 128) sx[i] = (sx[i] - mu) * r * scale[i];
  __syncthreads();
  float a = 0.f;
  for (int i = 0; i < 384; ++i) a += sx[i] * w[i * CA + tid];
  out[(long)t * CA + tid] = a;
}

// ---------------------------------------------------------------------------
// Kernel 3: queries_single_cond (S*Q, CA) = gather(tasc) + gather(tsc)
// ---------------------------------------------------------------------------
__global__ void __launch_bounds__(256)
k_qsc(const float* __restrict__ tasc, const float* __restrict__ tsc,
      const int* __restrict__ a2q_idx, const unsigned char* __restrict__ a2q_mask,
      const int* __restrict__ t2q_idx, const unsigned char* __restrict__ t2q_mask,
      float* __restrict__ o_qsc)
{
  const int gid = blockIdx.x * 256 + threadIdx.x;  // < S*Q*CA
  const int row = gid >> 7, c = gid & 127;
  float v = 0.f;
  if (a2q_mask[row]) v = tasc[(long)a2q_idx[row] * CA + c];
  if (t2q_mask[row]) v += tsc[(long)t2q_idx[row] * CA + c];
  o_qsc[gid] = v;
}

// ---------------------------------------------------------------------------
// Kernel 4: keys_single_cond (S*K, CA) = gather(qsc), keys_mask
// ---------------------------------------------------------------------------
__global__ void __launch_bounds__(256)
k_ksc(const float* __restrict__ qsc,
      const int* __restrict__ q2k_idx, const unsigned char* __restrict__ q2k_mask,
      const unsigned char* __restrict__ queries_mask,
      float* __restrict__ o_ksc, float* __restrict__ o_km)
{
  const int gid = blockIdx.x * 256 + threadIdx.x;  // < S*K*CA
  const int row = gid >> 7, c = gid & 127;
  const int  qi = q2k_idx[row];
  const bool qm = q2k_mask[row] != 0;
  o_ksc[gid] = qm ? qsc[(long)qi * CA + c] : 0.f;
  if (c == 0) o_km[row] = (qm && queries_mask[qi]) ? 1.f : 0.f;
}

// ---------------------------------------------------------------------------
// Kernel 5: q_pos/q_uid/k_pos/k_uid (resolve the q->k double indirection here)
// ---------------------------------------------------------------------------
__global__ void __launch_bounds__(256)
k_geom(const float* __restrict__ ref_ops, const int* __restrict__ ref_uid,
       const int* __restrict__ a2q_idx, const unsigned char* __restrict__ a2q_mask,
       const int* __restrict__ q2k_idx, const unsigned char* __restrict__ q2k_mask,
       float* __restrict__ q_pos, int* __restrict__ q_uid,
       float* __restrict__ k_pos, int* __restrict__ k_uid)
{
  const int i = blockIdx.x * 256 + threadIdx.x;  // < S*K = 73728
  if (i < S_SUB * Q_NUM) {
    const int  ai = a2q_idx[i];
    const bool am = a2q_mask[i] != 0;
    q_pos[i * 3 + 0] = am ? ref_ops[ai * 3 + 0] : 0.f;
    q_pos[i * 3 + 1] = am ? ref_ops[ai * 3 + 1] : 0.f;
    q_pos[i * 3 + 2] = am ? ref_ops[ai * 3 + 2] : 0.f;
    q_uid[i] = am ? ref_uid[ai] : 0;
  }
  const int  qi = q2k_idx[i];
  const bool qm = q2k_mask[i] != 0;
  const int  ai2 = a2q_idx[qi];
  const bool ok  = qm && (a2q_mask[qi] != 0);
  k_pos[i * 3 + 0] = ok ? ref_ops[ai2 * 3 + 0] : 0.f;
  k_pos[i * 3 + 1] = ok ? ref_ops[ai2 * 3 + 1] : 0.f;
  k_pos[i * 3 + 2] = ok ? ref_ops[ai2 * 3 + 2] : 0.f;
  k_uid[i] = qm ? ref_uid[qi] : 0;   // raw ref_space_uid gathered by q->k idx
}

// ---------------------------------------------------------------------------
// Kernel 6a: fold LN scale into pair projection weight: w_tp2 = diag(s) @ w_tp
// ---------------------------------------------------------------------------
__global__ void __launch_bounds__(256)
k_fold(const float* __restrict__ ln_p, const float* __restrict__ w_tp,
       float* __restrict__ w_tp2)
{
  const int i = blockIdx.x * 256 + threadIdx.x;  // < 128*16
  w_tp2[i] = ln_p[i >> 4] * w_tp[i];
}

// ---------------------------------------------------------------------------
// Kernel 6b: tpc = LN(trunk_pair_cond) @ w_tp  -> (T*T, 16).
// One wave per 16-row tile (rows contiguous in memory -> 8KB block).
// Async global->LDS staging, shuffle-free-ish LN stats, WMMA projection.
// ---------------------------------------------------------------------------
__global__ void __launch_bounds__(256)
k_tpc(const float* __restrict__ trunk_pair, const float* __restrict__ w_tp2,
      float* __restrict__ tpc)
{
  __shared__ float tile[8][16 * TPC_PAD];   // 8448B per wave, 66KB per block
  const int wave = threadIdx.x >> 5, lane = threadIdx.x & 31;
  const int m = lane & 15, h = lane >> 4;
  const long t16 = (long)blockIdx.x * 8 + wave;     // tile id, < 36864
  float* Tl = tile[wave];
  // Flat LDS addresses carry the LDS offset in addr[31:0] (ISA 10.2) ->
  // truncation yields the DS address for the async-LDS destination VGPR.
  const unsigned lds_base = (unsigned)(size_t)Tl;

  // Async-copy 16 rows of 512B each (row stride in LDS padded to 528B).
  const float* gsrc = trunk_pair + t16 * (16 * CA);
#pragma unroll
  for (int r = 0; r < 16; ++r) {
    unsigned long long ga = (unsigned long long)(size_t)(gsrc + r * CA) +
                            (unsigned long long)(lane * 16);
    unsigned la = lds_base + (unsigned)(r * TPC_PAD * 4 + lane * 16);
    asm volatile("global_load_async_to_lds_b128 %0, %1, off"
                 :: "v"(la), "v"(ga) : "memory");
  }
  asm volatile("s_wait_asynccnt 0x0" ::: "memory");

  // LN stats for row m: lane (m,h) scans columns [h*64, h*64+64).
  float s = 0.f, s2 = 0.f;
  const float* rowp = &Tl[m * TPC_PAD + h * 64];
#pragma unroll
  for (int i = 0; i < 16; ++i) {
    const float4 v = *(const float4*)(rowp + i * 4);
    s  += v.x + v.y + v.z + v.w;
    s2 += v.x * v.x + v.y * v.y + v.z * v.z + v.w * v.w;
  }
  s  += __shfl_xor(s, 16, 32);
  s2 += __shfl_xor(s2, 16, 32);
  const float mu  = s * (1.f / 128.f);
  const float var = s2 * (1.f / 128.f) - mu * mu;
  const float rr  = rsqrtf(var + 1e-5f);

  // Preload B fragments of w_tp2 (lane = N = m, K = 4j + 2h + r).
  v2f b[32];
#pragma unroll
  for (int j = 0; j < 32; ++j) {
    const int k = 4 * j + 2 * h;
    b[j].x = w_tp2[k * CP + m];
    b[j].y = w_tp2[(k + 1) * CP + m];
  }

  // D(16x16) = LN(rows) @ w_tp2 via 32 x WMMA f32 16x16x4.
  v8f acc = {};
#pragma unroll
  for (int j = 0; j < 32; ++j) {
    const int k = 4 * j + 2 * h;
    v2f a;
    a.x = (Tl[m * TPC_PAD + k]     - mu) * rr;
    a.y = (Tl[m * TPC_PAD + k + 1] - mu) * rr;
    acc = wmma_f32_4(a, b[j], acc);
  }

  float* o = tpc + t16 * (16 * CP);
#pragma unroll
  for (int v = 0; v < 8; ++v) o[(v + 8 * h) * CP + m] = acc[v];
}

// ---------------------------------------------------------------------------
// Kernel 7: Out(M,16) = relu(X(M,128)) @ W(128,16) via WMMA f32 16x16x4.
// ---------------------------------------------------------------------------
__global__ void __launch_bounds__(256)
k_proj(const float* __restrict__ X, const float* __restrict__ W,
       float* __restrict__ Out)
{
  const int wave = threadIdx.x >> 5, lane = threadIdx.x & 31;
  const int n = lane & 15, h = lane >> 4;
  const int tile = blockIdx.x * 8 + wave;
  const float* xr = X + (long)(tile * 16 + n) * CA;
  v8f acc = {};
#pragma unroll 8
  for (int k0 = 0; k0 < CA; k0 += 4) {
    const int k = k0 + 2 * h;
    v2f a; a.x = fmaxf(xr[k], 0.f);       a.y = fmaxf(xr[k + 1], 0.f);
    v2f b; b.x = W[k * CP + n];           b.y = W[(k + 1) * CP + n];
    acc = wmma_f32_4(a, b, acc);
  }
  float* o = Out + (long)tile * 256;      // 16 rows x 16 ch
#pragma unroll
  for (int v = 0; v < 8; ++v) o[(v + 8 * h) * CP + n] = acc[v];
}

// ---------------------------------------------------------------------------
// Kernel 8: fused pair kernel. One wave per 16-position (s,q,kt) tile:
// pair = row+col+tpc_gather+geometry, then 3 WMMA MLP stages with LDS
// transpose (C-layout -> A-layout) between stages; residual via WMMA C input.
// ---------------------------------------------------------------------------
__global__ void __launch_bounds__(256)
k_pair(const float* __restrict__ row_act, const float* __restrict__ col_act,
       const float* __restrict__ tpc,
       const float* __restrict__ q_pos, const float* __restrict__ k_pos,
       const int* __restrict__ q_uid, const int* __restrict__ k_uid,
       const int* __restrict__ t2q_idx, const unsigned char* __restrict__ t2q_mask,
       const int* __restrict__ t2k_idx, const unsigned char* __restrict__ t2k_mask,
       const float* __restrict__ w_off, const float* __restrict__ w_dist,
       const float* __restrict__ w_valid,
       const float* __restrict__ w_m1, const float* __restrict__ w_m2,
       const float* __restrict__ w_m3,
       float* __restrict__ out_pair)
{
  __shared__ float lds[8][16 * 17];   // per-wave padded transpose tile
  const int wave = threadIdx.x >> 5;
  const int lane = threadIdx.x & 31;
  const int c = lane & 15;            // channel (N) in C/D layout
  const int h = lane >> 4;
  const int tid = blockIdx.x * 8 + wave;   // 0..147455
  const int kt = tid & 7;
  const int q  = (tid >> 3) & 31;
  const int s  = tid >> 8;
  float* Tl = lds[wave];

  // Preload MLP B fragments (lane=N=c, K = 4j + 2h + r)
  v2f b1[4], b2[4], b3[4];
#pragma unroll
  for (int j = 0; j < 4; ++j) {
    const int k = 4 * j + 2 * h;
    b1[j].x = w_m1[k * CP + c]; b1[j].y = w_m1[(k + 1) * CP + c];
    b2[j].x = w_m2[k * CP + c]; b2[j].y = w_m2[(k + 1) * CP + c];
    b3[j].x = w_m3[k * CP + c]; b3[j].y = w_m3[(k + 1) * CP + c];
  }
  const float wo0 = w_off[0 * CP + c], wo1 = w_off[1 * CP + c], wo2 = w_off[2 * CP + c];
  const float wdc = w_dist[c], wvc = w_valid[c];

  const int   qrow = s * Q_NUM + q;
  const float ra   = row_act[(long)qrow * CP + c];
  const int   tq   = t2q_idx[qrow];
  const bool  mq   = t2q_mask[qrow] != 0;
  const float qp0 = q_pos[qrow * 3 + 0], qp1 = q_pos[qrow * 3 + 1], qp2 = q_pos[qrow * 3 + 2];
  const int   qu  = q_uid[qrow];

  float acc[8];
#pragma unroll
  for (int v = 0; v < 8; ++v) {
    const int kk   = kt * 16 + v + 8 * h;          // M index -> k position
    const int krow = s * K_NUM + kk;
    float val = ra + col_act[(long)krow * CP + c];
    const int tk = t2k_idx[krow];
    if (mq && (t2k_mask[krow] != 0))
      val += tpc[((long)tq * T_TOK + tk) * CP + c];
    const float o0 = qp0 - k_pos[krow * 3 + 0];
    const float o1 = qp1 - k_pos[krow * 3 + 1];
    const float o2 = qp2 - k_pos[krow * 3 + 2];
    if (qu == k_uid[krow]) {
      const float sq = o0 * o0 + o1 * o1 + o2 * o2;
      val += o0 * wo0 + o1 * wo1 + o2 * wo2 + wdc / (1.f + sq) + wvc;
    }
    acc[v] = val;
  }

  v8f pc;
#pragma unroll
  for (int v = 0; v < 8; ++v) pc[v] = acc[v];

  const int m = c;  // A-matrix row index for loads (lane&15)

  // ---- stage 1: d1 = relu(pair) @ w_m1
  __syncthreads();
#pragma unroll
  for (int v = 0; v < 8; ++v) Tl[(v + 8 * h) * 17 + c] = fmaxf(acc[v], 0.f);
  __syncthreads();
  v8f d1 = {};
#pragma unroll
  for (int j = 0; j < 4; ++j) {
    v2f a; a.x = Tl[m * 17 + 4 * j + 2 * h]; a.y = Tl[m * 17 + 4 * j + 2 * h + 1];
    d1 = wmma_f32_4(a, b1[j], d1);
  }
  // ---- stage 2: d2 = relu(d1) @ w_m2
  __syncthreads();
#pragma unroll
  for (int v = 0; v < 8; ++v) Tl[(v + 8 * h) * 17 + c] = fmaxf(d1[v], 0.f);
  __syncthreads();
  v8f d2 = {};
#pragma unroll
  for (int j = 0; j < 4; ++j) {
    v2f a; a.x = Tl[m * 17 + 4 * j + 2 * h]; a.y = Tl[m * 17 + 4 * j + 2 * h + 1];
    d2 = wmma_f32_4(a, b2[j], d2);
  }
  // ---- stage 3: out = pair + relu(d2) @ w_m3   (residual as WMMA C input)
  __syncthreads();
#pragma unroll
  for (int v = 0; v < 8; ++v) Tl[(v + 8 * h) * 17 + c] = fmaxf(d2[v], 0.f);
  __syncthreads();
  v8f d3 = pc;
#pragma unroll
  for (int j = 0; j < 4; ++j) {
    v2f a; a.x = Tl[m * 17 + 4 * j + 2 * h]; a.y = Tl[m * 17 + 4 * j + 2 * h + 1];
    d3 = wmma_f32_4(a, b3[j], d3);
  }

  float* po = out_pair + ((long)qrow * K_NUM + kt * 16) * CP;
#pragma unroll
  for (int v = 0; v < 8; ++v) po[(v + 8 * h) * CP + c] = d3[v];
}

// ---------------------------------------------------------------------------
// Host launcher
// ---------------------------------------------------------------------------
extern "C" void kernel_launch(void* const* d_in, const int* in_sizes, int n_in,
                              void* d_out, int out_size, void* d_ws, size_t ws_size,
                              hipStream_t stream)
{
  (void)in_sizes; (void)n_in; (void)out_size; (void)ws_size;
  const float* trunk_single = (const float*)d_in[0];
  const float* trunk_pair   = (const float*)d_in[1];
  const float* ref_ops      = (const float*)d_in[2];
  const float* ref_mask     = (const float*)d_in[3];
  const int*   ref_elem     = (const int*)d_in[4];
  const float* ref_chg      = (const float*)d_in[5];
  const int*   ref_name     = (const int*)d_in[6];
  const int*   ref_uid      = (const int*)d_in[7];
  const unsigned char* queries_mask = (const unsigned char*)d_in[8];
  const int*           a2q_idx  = (const int*)d_in[9];
  const unsigned char* a2q_mask = (const unsigned char*)d_in[10];
  const int*           q2k_idx  = (const int*)d_in[11];
  const unsigned char* q2k_mask = (const unsigned char*)d_in[12];
  const int*           t2q_idx  = (const int*)d_in[13];
  const unsigned char* t2q_mask = (const unsigned char*)d_in[14];
  const int*           t2k_idx  = (const int*)d_in[15];
  const unsigned char* t2k_mask = (const unsigned char*)d_in[16];
  const float* w_pos   = (const float*)d_in[17];
  const float* w_maskw = (const float*)d_in[18];
  const float* w_elem  = (const float*)d_in[19];
  const float* w_chg   = (const float*)d_in[20];
  const float* w_name  = (const float*)d_in[21];
  const float* ln_s    = (const float*)d_in[22];
  const float* w_ts    = (const float*)d_in[23];
  const float* w_row   = (const float*)d_in[24];
  const float* w_col   = (const float*)d_in[25];
  const float* ln_p    = (const float*)d_in[26];
  const float* w_tp    = (const float*)d_in[27];
  const float* w_off   = (const float*)d_in[28];
  const float* w_dist  = (const float*)d_in[29];
  const float* w_valid = (const float*)d_in[30];
  const float* w_m1    = (const float*)d_in[31];
  const float* w_m2    = (const float*)d_in[32];
  const float* w_m3    = (const float*)d_in[33];

  // Output layout (tuple order, flat f32)
  float* out    = (float*)d_out;
  float* o_qsc  = out;                         // (S,Q,CA)   2,359,296
  float* o_pair = out + 2359296;               // (S,Q,K,CP) 37,748,736
  float* o_km   = out + 2359296 + 37748736;    // (S,K)      73,728
  float* o_ksc  = o_km + 73728;                // (S,K,CA)   9,437,184

  // Workspace layout (floats)
  float* ws      = (float*)d_ws;
  float* tasc    = ws;                     // (T*A, CA)   2,359,296
  float* tsc     = ws + 2359296;           // (T, CA)        98,304
  float* row_act = ws + 2457600;           // (S*Q, CP)     294,912
  float* col_act = ws + 2752512;           // (S*K, CP)   1,179,648
  float* tpc     = ws + 3932160;           // (T*T, CP)   9,437,184
  float* q_pos   = ws + 13369344;          // (S*Q, 3)       55,296
  float* k_pos   = ws + 13424640;          // (S*K, 3)      221,184
  int*   q_uid   = (int*)(ws + 13645824);  // (S*Q)          18,432
  int*   k_uid   = (int*)(ws + 13664256);  // (S*K)          73,728
  float* w_tp2   = ws + 13737984;          // (128,16)        2,048

  // 1) atom conditioning
  k_atom<<<T_TOK * A_ATM, 128, 0, stream>>>(ref_ops, ref_mask, ref_elem, ref_chg,
                                            ref_name, w_pos, w_maskw, w_elem,
                                            w_chg, w_name, tasc);
  // 2) trunk single LN+proj
  k_tsc<<<T_TOK, 128, 0, stream>>>(trunk_single, ln_s, w_ts, tsc);
  // 3) queries_single_cond (also an output)
  k_qsc<<<(S_SUB * Q_NUM * CA) / 256, 256, 0, stream>>>(tasc, tsc, a2q_idx, a2q_mask,
                                                        t2q_idx, t2q_mask, o_qsc);
  // 4) keys_single_cond + keys_mask (outputs)
  k_ksc<<<(S_SUB * K_NUM * CA) / 256, 256, 0, stream>>>(o_qsc, q2k_idx, q2k_mask,
                                                        queries_mask, o_ksc, o_km);
  // 5) geometry gathers
  k_geom<<<(S_SUB * K_NUM) / 256, 256, 0, stream>>>(ref_ops, ref_uid, a2q_idx,
                                                    a2q_mask, q2k_idx, q2k_mask,
                                                    q_pos, q_uid, k_pos, k_uid);
  // 6) trunk pair LN+proj (big HBM pass; WMMA + async global->LDS)
  k_fold<<<(CA * CP) / 256, 256, 0, stream>>>(ln_p, w_tp, w_tp2);
  k_tpc<<<(T_TOK * T_TOK / 16) / 8, 256, 0, stream>>>(trunk_pair, w_tp2, tpc);
  // 7) row/col projections via WMMA
  k_proj<<<(S_SUB * Q_NUM / 16) / 8, 256, 0, stream>>>(o_qsc, w_row, row_act);
  k_proj<<<(S_SUB * K_NUM / 16) / 8, 256, 0, stream>>>(o_ksc, w_col, col_act);
  // 8) fused pair assembly + WMMA MLP
  k_pair<<<(S_SUB * Q_NUM * (K_NUM / 16)) / 8, 256, 0, stream>>>(
      row_act, col_act, tpc, q_pos, k_pos, q_uid, k_uid,
      t2q_idx, t2q_mask, t2k_idx, t2k_mask,
      w_off, w_dist, w_valid, w_m1, w_m2, w_m3, o_pair);
}